// Holo_65515431133936
// MI455X (gfx1250) — compile-verified
//
#include <hip/hip_runtime.h>

#define NV      12000
#define NC      4000
#define NN      16000
#define NE      96000
#define GG      4
#define NVP     3000
#define NCP     1000
#define EPG     24000   // edges per graph
#define DEGV    8
#define DF      64
#define DH      128
#define DOUT    64
#define KB      8
#define MAXSET  512

typedef float v2f __attribute__((ext_vector_type(2)));
typedef float v8f __attribute__((ext_vector_type(8)));

// ---------------- adjacency construction ----------------

__global__ void k_count(const int* __restrict__ econ, int* __restrict__ cnt) {
  int e = blockIdx.x * blockDim.x + threadIdx.x;
  if (e < NE) atomicAdd(&cnt[econ[e]], 1);
}

__global__ void k_offsets(const int* __restrict__ cnt, int* __restrict__ off) {
  if (blockIdx.x == 0 && threadIdx.x == 0) {
    int acc = 0;
    off[0] = 0;
    for (int c = 0; c < NC; ++c) { acc += cnt[c]; off[c + 1] = acc; }
  }
}

// deterministic fill: each con scans its graph's edge range in order
__global__ void k_fill(const int* __restrict__ econ, const int* __restrict__ off,
                       int* __restrict__ eid) {
  int c = blockIdx.x * blockDim.x + threadIdx.x;
  if (c >= NC) return;
  int g = c / NCP;
  int base = off[c], k = 0;
  int e0 = g * EPG, e1 = e0 + EPG;
  for (int e = e0; e < e1; ++e)
    if (econ[e] == c) eid[base + (k++)] = e;
}

__global__ void k_dinv(const int* __restrict__ off, float* __restrict__ dinv,
                       float* __restrict__ loopv) {
  int n = blockIdx.x * blockDim.x + threadIdx.x;
  if (n >= NN) return;
  int cnt = (n < NV) ? DEGV : (off[n - NV + 1] - off[n - NV]);
  float d = (float)(cnt + 1);
  float di = rsqrtf(d);
  dinv[n] = di;
  loopv[n] = di * di;
}

__global__ void k_ev(const int* __restrict__ econ, const int* __restrict__ evar,
                     const float* __restrict__ dinv, float* __restrict__ ev) {
  int e = blockIdx.x * blockDim.x + threadIdx.x;
  if (e < NE) ev[e] = dinv[evar[e]] * dinv[NV + econ[e]];
}

__global__ void k_nodedeg(const int* __restrict__ econ, const int* __restrict__ off,
                          const int* __restrict__ eid, const float* __restrict__ ev,
                          const float* __restrict__ loopv, float* __restrict__ nd) {
  int n = blockIdx.x * blockDim.x + threadIdx.x;
  if (n >= NN) return;
  float s = loopv[n];
  if (n < NV) {
    int e0 = n * DEGV;
    #pragma unroll
    for (int j = 0; j < DEGV; ++j) s += ev[e0 + j];
  } else {
    int c = n - NV;
    for (int p = off[c]; p < off[c + 1]; ++p) s += ev[eid[p]];
  }
  nd[n] = s;
}

// top-8 weighted degree per graph, stable tie-break (lower sample index wins)
__global__ void k_topk(const float* __restrict__ nd, int* __restrict__ bidx) {
  __shared__ float vals[NVP + NCP];
  __shared__ float rv[256];
  __shared__ int   ri[256];
  int g = blockIdx.x, tid = threadIdx.x;
  for (int s = tid; s < NVP + NCP; s += 256) {
    int node = (s < NVP) ? g * NVP + s : NV + g * NCP + (s - NVP);
    vals[s] = nd[node];
  }
  __syncthreads();
  for (int it = 0; it < KB; ++it) {
    float bv = -1e30f; int bs = 1 << 30;
    for (int s = tid; s < NVP + NCP; s += 256) {
      float v = vals[s];
      if (v > bv || (v == bv && s < bs)) { bv = v; bs = s; }
    }
    rv[tid] = bv; ri[tid] = bs;
    __syncthreads();
    for (int o = 128; o > 0; o >>= 1) {
      if (tid < o) {
        if (rv[tid + o] > rv[tid] ||
            (rv[tid + o] == rv[tid] && ri[tid + o] < ri[tid])) {
          rv[tid] = rv[tid + o]; ri[tid] = ri[tid + o];
        }
      }
      __syncthreads();
    }
    if (tid == 0) {
      int s = ri[0];
      bidx[g * KB + it] = (s < NVP) ? g * NVP + s : NV + g * NCP + (s - NVP);
      vals[s] = -2e30f;
    }
    __syncthreads();
  }
}

// ---------------- WMMA GEMMs (V_WMMA_F32_16X16X4_F32) ----------------
// one wave per 16x16 C tile; ISA layouts: A lane<16 -> K{0,1}, lane>=16 -> K{2,3};
// B/C/D row-striped across lanes.

__global__ void k_gemm1(const float* __restrict__ vf, const float* __restrict__ cf,
                        const float* __restrict__ W1, const float* __restrict__ b1,
                        float* __restrict__ Z0) {
  int wave = (blockIdx.x * blockDim.x + threadIdx.x) >> 5;
  if (wave >= (NN / 16) * (DH / 16)) return;
  int tm = wave >> 3, tn = wave & 7;
  int lane = threadIdx.x & 31, half = lane >> 4, l16 = lane & 15;
  int row = tm * 16 + l16;
  const float* Arow = (row < NV) ? (vf + (size_t)row * DF)
                                 : (cf + (size_t)(row - NV) * DF);
  int col = tn * 16 + l16;
  v8f acc = {};
  #pragma unroll
  for (int k = 0; k < DF / 4; ++k) {
    int kk = k * 4 + half * 2;
    v2f a = *(const v2f*)(Arow + kk);
    v2f b;
    b.x = W1[kk * DH + col];
    b.y = W1[(kk + 1) * DH + col];
    acc = __builtin_amdgcn_wmma_f32_16x16x4_f32(false, a, false, b,
                                                (short)0, acc, false, false);
  }
  float bb = b1[col];
  #pragma unroll
  for (int i = 0; i < 8; ++i) {
    int r = tm * 16 + half * 8 + i;
    Z0[(size_t)r * DH + col] = acc[i] + bb;
  }
}

__global__ void k_gemm2(const float* __restrict__ S0, const float* __restrict__ W2,
                        const float* __restrict__ b2, float* __restrict__ P0) {
  int wave = (blockIdx.x * blockDim.x + threadIdx.x) >> 5;
  if (wave >= (NN / 16) * (DOUT / 16)) return;
  int tm = wave >> 2, tn = wave & 3;
  int lane = threadIdx.x & 31, half = lane >> 4, l16 = lane & 15;
  int row = tm * 16 + l16;
  int col = tn * 16 + l16;
  v8f acc = {};
  #pragma unroll
  for (int k = 0; k < DH / 4; ++k) {
    int kk = k * 4 + half * 2;
    v2f a = *(const v2f*)(S0 + (size_t)row * DH + kk);
    a.x = fmaxf(a.x, 0.f);                 // fused relu(S0)
    a.y = fmaxf(a.y, 0.f);
    v2f b;
    b.x = W2[kk * DOUT + col];
    b.y = W2[(kk + 1) * DOUT + col];
    acc = __builtin_amdgcn_wmma_f32_16x16x4_f32(false, a, false, b,
                                                (short)0, acc, false, false);
  }
  float bb = b2[col];
  #pragma unroll
  for (int i = 0; i < 8; ++i) {
    int r = tm * 16 + half * 8 + i;
    P0[(size_t)r * DOUT + col] = acc[i] + bb;
  }
}

// ---------------- gather-based spmm (deterministic) ----------------

__global__ void k_spmm(const float* __restrict__ in, float* __restrict__ out,
                       const float* __restrict__ ev, const float* __restrict__ loopv,
                       const int* __restrict__ econ, const int* __restrict__ evar,
                       const int* __restrict__ off, const int* __restrict__ eid,
                       int fsh) {
  int idx = blockIdx.x * blockDim.x + threadIdx.x;
  if (idx >= (NN << fsh)) return;
  int n = idx >> fsh;
  int f = idx & ((1 << fsh) - 1);
  float acc = loopv[n] * in[idx];
  if (n < NV) {
    int e0 = n * DEGV;
    #pragma unroll
    for (int j = 0; j < DEGV; ++j) {
      int e = e0 + j;
      acc += ev[e] * in[((NV + econ[e]) << fsh) + f];
    }
  } else {
    int c = n - NV;
    for (int p = off[c]; p < off[c + 1]; ++p) {
      int e = eid[p];
      acc += ev[e] * in[(evar[e] << fsh) + f];
    }
  }
  out[idx] = acc;
}

// ---------------- per-break sparse correction ----------------

// build deduped neighbor set of break node t (a = column of A), write node->slot map
__global__ void k_buildset(const int* __restrict__ bidx, int b,
                           const int* __restrict__ econ, const int* __restrict__ evar,
                           const int* __restrict__ off, const int* __restrict__ eid,
                           const float* __restrict__ ev, const float* __restrict__ loopv,
                           int* __restrict__ setN, float* __restrict__ setA,
                           int* __restrict__ setLen, int* __restrict__ map) {
  if (threadIdx.x != 0 || blockIdx.x != 0) return;
  int t = bidx[b];
  int L = 0;
  setN[L] = t; setA[L] = loopv[t]; L++;
  if (t < NV) {
    int e0 = t * DEGV;
    for (int j = 0; j < DEGV; ++j) {
      int n = NV + econ[e0 + j];
      float a = ev[e0 + j];
      int found = -1;
      for (int i = 0; i < L; ++i) if (setN[i] == n) { found = i; break; }
      if (found >= 0) setA[found] += a;
      else if (L < MAXSET) { setN[L] = n; setA[L] = a; L++; }
    }
  } else {
    int c = t - NV;
    for (int p = off[c]; p < off[c + 1]; ++p) {
      int e = eid[p];
      int n = evar[e];
      float a = ev[e];
      int found = -1;
      for (int i = 0; i < L; ++i) if (setN[i] == n) { found = i; break; }
      if (found >= 0) setA[found] += a;
      else if (L < MAXSET) { setN[L] = n; setA[L] = a; L++; }
    }
  }
  *setLen = L;
  for (int i = 0; i < L; ++i) map[setN[i]] = i;   // map was memset to -1
}

// dP[i] = (relu(S0[n]+a*w1last) - relu(S0[n])) @ W2   for each set entry
__global__ void k_deltap(const int* __restrict__ setN, const float* __restrict__ setA,
                         const int* __restrict__ setLen,
                         const float* __restrict__ S0, const float* __restrict__ W1,
                         const float* __restrict__ W2, float* __restrict__ dP) {
  int i = blockIdx.x;
  if (i >= *setLen) return;
  __shared__ float dh[DH];
  int n = setN[i];
  float a = setA[i];
  int tid = threadIdx.x;
  float s = S0[(size_t)n * DH + tid];
  dh[tid] = fmaxf(s + a * W1[DF * DH + tid], 0.f) - fmaxf(s, 0.f);
  __syncthreads();
  if (tid < DOUT) {
    float acc = 0.f;
    for (int k = 0; k < DH; ++k) acc += dh[k] * W2[k * DOUT + tid];
    dP[i * DOUT + tid] = acc;
  }
}

// gather A*dP over this group's var rows, fuse LayerNorm and 1/8 accumulation
__global__ void k_accum(int g, const int* __restrict__ map, const float* __restrict__ dP,
                        const float* __restrict__ ev, const float* __restrict__ loopv,
                        const int* __restrict__ econ, const float* __restrict__ Y0,
                        const float* __restrict__ gamma, const float* __restrict__ beta,
                        float* __restrict__ out) {
  __shared__ float s1[DOUT];
  __shared__ float s2[DOUT];
  int v = g * NVP + blockIdx.x;
  int o = threadIdx.x;
  float delta = 0.f;
  int i0 = map[v];
  if (i0 >= 0) delta += loopv[v] * dP[i0 * DOUT + o];
  int e0 = v * DEGV;
  #pragma unroll
  for (int j = 0; j < DEGV; ++j) {
    int m = map[NV + econ[e0 + j]];
    if (m >= 0) delta += ev[e0 + j] * dP[m * DOUT + o];
  }
  float y = Y0[(size_t)v * DOUT + o] + delta;
  s1[o] = y; s2[o] = y * y;
  __syncthreads();
  for (int off = 32; off > 0; off >>= 1) {
    if (o < off) { s1[o] += s1[o + off]; s2[o] += s2[o + off]; }
    __syncthreads();
  }
  float mu = s1[0] * (1.f / DOUT);
  float va = s2[0] * (1.f / DOUT) - mu * mu;
  float val = (y - mu) * rsqrtf(va + 1e-5f) * gamma[o] + beta[o];
  out[(size_t)v * DOUT + o] += 0.125f * val;
}

// ---------------- driver ----------------

extern "C" void kernel_launch(void* const* d_in, const int* in_sizes, int n_in,
                              void* d_out, int out_size, void* d_ws, size_t ws_size,
                              hipStream_t stream) {
  const float* vf    = (const float*)d_in[0];
  const float* cf    = (const float*)d_in[1];
  const float* W1    = (const float*)d_in[2];
  const float* b1    = (const float*)d_in[3];
  const float* W2    = (const float*)d_in[4];
  const float* b2    = (const float*)d_in[5];
  const float* gamma = (const float*)d_in[6];
  const float* beta  = (const float*)d_in[7];
  const int*   econ  = (const int*)d_in[8];
  const int*   evar  = econ + NE;
  float* out = (float*)d_out;

  char* w = (char*)d_ws;
  size_t off = 0;
  auto grab = [&](size_t bytes) { void* p = w + off; off = (off + bytes + 255) & ~(size_t)255; return p; };
  float* Z0      = (float*)grab((size_t)NN * DH * 4);   // reused: P0 = Z0[0..], Y0 = Z0[NN*DOUT..]
  float* S0      = (float*)grab((size_t)NN * DH * 4);
  float* dinv    = (float*)grab(NN * 4);
  float* loopv   = (float*)grab(NN * 4);
  float* ev      = (float*)grab(NE * 4);
  float* nd      = (float*)grab(NN * 4);
  float* dP      = (float*)grab(MAXSET * DOUT * 4);
  float* setA    = (float*)grab(MAXSET * 4);
  int*   con_cnt = (int*)grab(NC * 4);
  int*   con_off = (int*)grab((NC + 1) * 4);
  int*   con_eid = (int*)grab(NE * 4);
  int*   setN    = (int*)grab(MAXSET * 4);
  int*   setLen  = (int*)grab(4);
  int*   bidx    = (int*)grab(GG * KB * 4);
  int*   nmap    = (int*)grab(NN * 4);
  if (off > ws_size) return;
  float* P0 = Z0;
  float* Y0 = Z0 + (size_t)NN * DOUT;

  hipMemsetAsync(out, 0, (size_t)NV * DOUT * 4, stream);
  hipMemsetAsync(con_cnt, 0, NC * 4, stream);

  k_count  <<<(NE + 255) / 256, 256, 0, stream>>>(econ, con_cnt);
  k_offsets<<<1, 1, 0, stream>>>(con_cnt, con_off);
  k_fill   <<<(NC + 255) / 256, 256, 0, stream>>>(econ, con_off, con_eid);
  k_dinv   <<<(NN + 255) / 256, 256, 0, stream>>>(con_off, dinv, loopv);
  k_ev     <<<(NE + 255) / 256, 256, 0, stream>>>(econ, evar, dinv, ev);
  k_nodedeg<<<(NN + 255) / 256, 256, 0, stream>>>(econ, con_off, con_eid, ev, loopv, nd);
  k_topk   <<<GG, 256, 0, stream>>>(nd, bidx);

  // shared dense pipeline (WMMA f32)
  k_gemm1<<<((NN / 16) * (DH / 16) * 32) / 128, 128, 0, stream>>>(vf, cf, W1, b1, Z0);
  k_spmm <<<((NN * DH) + 255) / 256, 256, 0, stream>>>(Z0, S0, ev, loopv, econ, evar, con_off, con_eid, 7);
  k_gemm2<<<((NN / 16) * (DOUT / 16) * 32) / 128, 128, 0, stream>>>(S0, W2, b2, P0);
  k_spmm <<<((NN * DOUT) + 255) / 256, 256, 0, stream>>>(P0, Y0, ev, loopv, econ, evar, con_off, con_eid, 6);

  // per-break sparse correction + LayerNorm + group mean
  for (int b = 0; b < GG * KB; ++b) {
    int g = b / KB;
    hipMemsetAsync(nmap, 0xFF, NN * 4, stream);   // -1
    k_buildset<<<1, 1, 0, stream>>>(bidx, b, econ, evar, con_off, con_eid,
                                    ev, loopv, setN, setA, setLen, nmap);
    k_deltap<<<MAXSET, DH, 0, stream>>>(setN, setA, setLen, S0, W1, W2, dP);
    k_accum <<<NVP, DOUT, 0, stream>>>(g, nmap, dP, ev, loopv, econ, Y0,
                                       gamma, beta, out);
  }
}